// FrameAveraging_54357106098670
// MI455X (gfx1250) — compile-verified
//
#include <hip/hip_runtime.h>

#define BATCH 512
#define NPTS  4096

typedef __attribute__((ext_vector_type(2))) float v2f;
typedef __attribute__((ext_vector_type(8))) float v8f;

// ---------------------------------------------------------------------------
// Kernel 1: per-batch stats.  One block per batch; 256 threads x 16 points.
// Accumulates: sum(m), sum(m^2), sum(m*X) [3], sum(X X^T) [6 upper tri]
// Then C_ij = SXX_ij - c_i SmX_j - c_j SmX_i + c_i c_j Sm2,  c = SmX / Sm.
// Writes ws_stats[b*9] = {C00,C01,C02,C11,C12,C22, cx,cy,cz}
// ---------------------------------------------------------------------------
__global__ __launch_bounds__(256) void fa_stats_kernel(
    const float* __restrict__ X, const float* __restrict__ M,
    float* __restrict__ ws_stats)
{
  const int b = blockIdx.x;
  const int t = threadIdx.x;
  const float* Xb = X + (size_t)b * NPTS * 3;
  const float* Mb = M + (size_t)b * NPTS;

  float sm = 0.f, sm2 = 0.f;
  float sx = 0.f, sy = 0.f, sz = 0.f;
  float xx = 0.f, xy = 0.f, xz = 0.f, yy = 0.f, yz = 0.f, zz = 0.f;

  for (int k = 0; k < 4; ++k) {
    const int p = k * 1024 + t * 4;                    // 4 consecutive points
    const float4* xv = (const float4*)(Xb + (size_t)p * 3);
    const float4 x0 = xv[0], x1 = xv[1], x2 = xv[2];   // 12 floats = 4 points
    const float4 mw = *(const float4*)(Mb + p);

    const float px[4] = {x0.x, x0.w, x1.z, x2.y};
    const float py[4] = {x0.y, x1.x, x1.w, x2.z};
    const float pz[4] = {x0.z, x1.y, x2.x, x2.w};
    const float pm[4] = {mw.x, mw.y, mw.z, mw.w};
#pragma unroll
    for (int i = 0; i < 4; ++i) {
      const float xm = px[i], ym = py[i], zm = pz[i], m = pm[i];
      sm += m;        sm2 += m * m;
      sx += m * xm;   sy += m * ym;   sz += m * zm;
      xx += xm * xm;  xy += xm * ym;  xz += xm * zm;
      yy += ym * ym;  yz += ym * zm;  zz += zm * zm;
    }
  }

  float vals[11] = {sm, sm2, sx, sy, sz, xx, xy, xz, yy, yz, zz};
  __shared__ float red[8 * 11];
  const int lane = t & 31, wave = t >> 5;
#pragma unroll
  for (int q = 0; q < 11; ++q) {
    float v = vals[q];
    for (int off = 16; off > 0; off >>= 1) v += __shfl_down(v, off, 32);
    if (lane == 0) red[wave * 11 + q] = v;
  }
  __syncthreads();
  if (t == 0) {
    float tot[11];
#pragma unroll
    for (int q = 0; q < 11; ++q) {
      float s = 0.f;
      for (int w = 0; w < 8; ++w) s += red[w * 11 + q];
      tot[q] = s;
    }
    const float Msum = tot[0], Msq = tot[1];
    const float invM = (Msum != 0.f) ? 1.f / Msum : 0.f;
    const float cx = tot[2] * invM, cy = tot[3] * invM, cz = tot[4] * invM;
    const float sxv = tot[2], syv = tot[3], szv = tot[4];
    float* o = ws_stats + b * 9;
    o[0] = tot[5]  - 2.f * cx * sxv + cx * cx * Msq;           // C00
    o[1] = tot[6]  - cx * syv - cy * sxv + cx * cy * Msq;      // C01
    o[2] = tot[7]  - cx * szv - cz * sxv + cx * cz * Msq;      // C02
    o[3] = tot[8]  - 2.f * cy * syv + cy * cy * Msq;           // C11
    o[4] = tot[9]  - cy * szv - cz * syv + cy * cz * Msq;      // C12
    o[5] = tot[10] - 2.f * cz * szv + cz * cz * Msq;           // C22
    o[6] = cx; o[7] = cy; o[8] = cz;
  }
}

// ---------------------------------------------------------------------------
// Kernel 2: closed-form symmetric 3x3 eigh per batch (1 thread / batch).
// Ascending eigenvalues; eigenvectors in columns of V (row-major V[i][j]).
// Writes ws_V[b*9], F_ops to output, center to output.
// ---------------------------------------------------------------------------
__device__ inline void crossp(const float* a, const float* c, float* o) {
  o[0] = a[1] * c[2] - a[2] * c[1];
  o[1] = a[2] * c[0] - a[0] * c[2];
  o[2] = a[0] * c[1] - a[1] * c[0];
}
__device__ inline float dot3(const float* a, const float* c) {
  return a[0] * c[0] + a[1] * c[1] + a[2] * c[2];
}
__device__ inline void eigvec(float c00, float c01, float c02, float c11,
                              float c12, float c22, float l, float* v) {
  const float r0[3] = {c00 - l, c01, c02};
  const float r1[3] = {c01, c11 - l, c12};
  const float r2[3] = {c02, c12, c22 - l};
  float cA[3], cB[3], cC[3];
  crossp(r0, r1, cA); crossp(r0, r2, cB); crossp(r1, r2, cC);
  const float nA = dot3(cA, cA), nB = dot3(cB, cB), nC = dot3(cC, cC);
  const float* best = cA; float nb = nA;
  if (nB > nb) { best = cB; nb = nB; }
  if (nC > nb) { best = cC; nb = nC; }
  if (nb < 1e-30f) { v[0] = 1.f; v[1] = 0.f; v[2] = 0.f; return; }
  const float inv = rsqrtf(nb);
  v[0] = best[0] * inv; v[1] = best[1] * inv; v[2] = best[2] * inv;
}

__global__ __launch_bounds__(256) void fa_eig_kernel(
    const float* __restrict__ ws_stats, float* __restrict__ ws_V,
    float* __restrict__ outF, float* __restrict__ outC)
{
  const int b = blockIdx.x * blockDim.x + threadIdx.x;
  if (b >= BATCH) return;
  const float* s = ws_stats + b * 9;
  const float c00 = s[0], c01 = s[1], c02 = s[2];
  const float c11 = s[3], c12 = s[4], c22 = s[5];

  const float p1 = c01 * c01 + c02 * c02 + c12 * c12;
  const float q  = (c00 + c11 + c22) * (1.f / 3.f);
  const float d0 = c00 - q, d1 = c11 - q, d2 = c22 - q;
  const float p2 = d0 * d0 + d1 * d1 + d2 * d2 + 2.f * p1;
  const float p  = sqrtf(p2 * (1.f / 6.f));

  float l0, l1, l2;
  float v0[3], v1[3], v2[3];
  if (p > 1e-20f) {
    const float ip  = 1.f / p;
    const float b00 = d0 * ip, b11 = d1 * ip, b22 = d2 * ip;
    const float b01 = c01 * ip, b02 = c02 * ip, b12 = c12 * ip;
    float r = 0.5f * (b00 * (b11 * b22 - b12 * b12)
                    - b01 * (b01 * b22 - b12 * b02)
                    + b02 * (b01 * b12 - b11 * b02));
    r = fminf(1.f, fmaxf(-1.f, r));
    const float phi = acosf(r) * (1.f / 3.f);
    l2 = q + 2.f * p * cosf(phi);                     // largest
    l0 = q + 2.f * p * cosf(phi + 2.0943951023931953f); // smallest
    l1 = 3.f * q - l0 - l2;
    eigvec(c00, c01, c02, c11, c12, c22, l0, v0);
    eigvec(c00, c01, c02, c11, c12, c22, l2, v2);
    crossp(v2, v0, v1);
    const float n1 = dot3(v1, v1);
    if (n1 > 1e-30f) {
      const float inv = rsqrtf(n1);
      v1[0] *= inv; v1[1] *= inv; v1[2] *= inv;
    } else { v1[0] = 0.f; v1[1] = 1.f; v1[2] = 0.f; }
  } else {
    l0 = l1 = l2 = q;
    v0[0] = 1.f; v0[1] = 0.f; v0[2] = 0.f;
    v1[0] = 0.f; v1[1] = 1.f; v1[2] = 0.f;
    v2[0] = 0.f; v2[1] = 0.f; v2[2] = 1.f;
  }
  (void)l1;

  // V row-major: V[i][j] = j-th eigenvector, i-th component
  float V[9];
#pragma unroll
  for (int i = 0; i < 3; ++i) {
    V[i * 3 + 0] = v0[i];
    V[i * 3 + 1] = v1[i];
    V[i * 3 + 2] = v2[i];
  }
  float* vw = ws_V + b * 9;
#pragma unroll
  for (int i = 0; i < 9; ++i) vw[i] = V[i];

  // F_ops[b,o,i,j] = OPS[o][j] * V[i][j];  OPS sign for axis a: +1 if bit(2-a) of o set
  float* F = outF + (size_t)b * 72;
#pragma unroll
  for (int o = 0; o < 8; ++o) {
    const float sj[3] = {((o >> 2) & 1) ? 1.f : -1.f,
                         ((o >> 1) & 1) ? 1.f : -1.f,
                         ((o >> 0) & 1) ? 1.f : -1.f};
#pragma unroll
    for (int i = 0; i < 3; ++i)
#pragma unroll
      for (int j = 0; j < 3; ++j)
        F[(o * 3 + i) * 3 + j] = sj[j] * V[i * 3 + j];
  }
  outC[b * 3 + 0] = s[6];
  outC[b * 3 + 1] = s[7];
  outC[b * 3 + 2] = s[8];
}

// ---------------------------------------------------------------------------
// Kernel 3: h[b,o,p,i] = OPS[o][i] * (Xc[p] . V[:,i])
// Projection y = Xc * V via V_WMMA_F32_16X16X4_F32 (A = V^T padded, per-batch
// constant; B = 16 centered points per issue). WMMA D rows 0..2 -> LDS
// transpose -> 8 sign-flipped float4 (b128) coalesced stores per thread.
// Block = 256 thr (8 waves), 1024 points; grid = (N/1024, B).
// ---------------------------------------------------------------------------
__device__ inline float fsx(float v, unsigned m) {
  return __uint_as_float(__float_as_uint(v) ^ m);
}

__global__ __launch_bounds__(256) void fa_proj_kernel(
    const float* __restrict__ X, const float* __restrict__ ws_stats,
    const float* __restrict__ ws_V, float* __restrict__ h)
{
  __shared__ float ytile[1024 * 3];       // 12 KB: y for the block's points

  const int b        = blockIdx.y;
  const int tileBase = blockIdx.x * 1024; // point base of this block
  const int t        = threadIdx.x;
  const int lane     = t & 31;
  const int wave     = t >> 5;
  const int lm       = lane & 15;
  const bool lo      = lane < 16;

  const float* Xb = X + (size_t)b * NPTS * 3;
  const float* st = ws_stats + b * 9;
  const float* Vw = ws_V + b * 9;         // row-major V[i*3+j]
  const float cx = st[6], cy = st[7], cz = st[8];

  // A (16x4 f32): A[m][k] = V[k][m] for m<3,k<3; zero elsewhere.
  // Layout: lanes 0-15 hold {K0,K1}, lanes 16-31 hold {K2,K3}.
  v2f a; a.x = 0.f; a.y = 0.f;
  if (lo) {
    if (lm < 3) { a.x = Vw[0 * 3 + lm]; a.y = Vw[1 * 3 + lm]; }
  } else {
    if (lm < 3) { a.x = Vw[2 * 3 + lm]; }  // K=3 row is zero padding
  }

  const float csel0 = lo ? cx : cz;
  const int   comp0 = lo ? 0 : 2;
  const int waveBase = wave * 128;        // 128 points per wave

  for (int s = 0; s < 8; ++s) {
    const int pl = waveBase + s * 16 + lm;   // point within block tile
    const int p  = tileBase + pl;
    // B (4x16 f32): VGPR0 = {row K0 | row K2}, VGPR1 = {row K1 | zero row K3}
    const float r0 = Xb[(size_t)p * 3 + comp0] - csel0;
    const float r1raw = Xb[(size_t)p * 3 + 1];
    const float r1 = lo ? (r1raw - cy) : 0.f;
    v2f bm; bm.x = r0; bm.y = r1;

    v8f acc = {0.f, 0.f, 0.f, 0.f, 0.f, 0.f, 0.f, 0.f};
    acc = __builtin_amdgcn_wmma_f32_16x16x4_f32(
        false, a, false, bm, (short)0, acc, false, false);

    if (lo) {                              // D rows 0..2, lanes 0..15 = y
      const int yo = pl * 3;
      ytile[yo + 0] = acc[0];
      ytile[yo + 1] = acc[1];
      ytile[yo + 2] = acc[2];
    }
  }
  __syncthreads();

  // Phase 2: 4 points / thread, 8 sign variants, b128 stores.
  const float4* ysrc = (const float4*)&ytile[t * 12];
  const float4 q0 = ysrc[0], q1 = ysrc[1], q2 = ysrc[2];
  const int pg = tileBase + t * 4;

#pragma unroll
  for (int o = 0; o < 8; ++o) {
    const unsigned m0 = ((o >> 2) & 1) ? 0u : 0x80000000u;
    const unsigned m1 = ((o >> 1) & 1) ? 0u : 0x80000000u;
    const unsigned m2 = ((o >> 0) & 1) ? 0u : 0x80000000u;
    // slot axis pattern over 12 contiguous floats: 0,1,2,0 | 1,2,0,1 | 2,0,1,2
    const float4 w0 = make_float4(fsx(q0.x, m0), fsx(q0.y, m1), fsx(q0.z, m2), fsx(q0.w, m0));
    const float4 w1 = make_float4(fsx(q1.x, m1), fsx(q1.y, m2), fsx(q1.z, m0), fsx(q1.w, m1));
    const float4 w2 = make_float4(fsx(q2.x, m2), fsx(q2.y, m0), fsx(q2.z, m1), fsx(q2.w, m2));
    float4* dst = (float4*)(h + ((size_t)(b * 8 + o) * NPTS + pg) * 3);
    dst[0] = w0; dst[1] = w1; dst[2] = w2;
  }
}

// ---------------------------------------------------------------------------
extern "C" void kernel_launch(void* const* d_in, const int* in_sizes, int n_in,
                              void* d_out, int out_size, void* d_ws, size_t ws_size,
                              hipStream_t stream) {
  (void)in_sizes; (void)n_in; (void)out_size; (void)ws_size;
  const float* X    = (const float*)d_in[0];
  const float* mask = (const float*)d_in[1];

  float* out  = (float*)d_out;
  float* h    = out;                                        // [B*8, N, 3]
  float* outF = out + (size_t)BATCH * 8 * NPTS * 3;         // [B, 8, 3, 3]
  float* outC = outF + (size_t)BATCH * 72;                  // [B, 3]

  float* ws_stats = (float*)d_ws;          // BATCH*9: C(6) + center(3)
  float* ws_V     = ws_stats + BATCH * 9;  // BATCH*9: V row-major

  fa_stats_kernel<<<BATCH, 256, 0, stream>>>(X, mask, ws_stats);
  fa_eig_kernel<<<(BATCH + 255) / 256, 256, 0, stream>>>(ws_stats, ws_V, outF, outC);
  fa_proj_kernel<<<dim3(NPTS / 1024, BATCH), 256, 0, stream>>>(X, ws_stats, ws_V, h);
}